// NRBS_9552007266807
// MI455X (gfx1250) — compile-verified
//
#include <hip/hip_runtime.h>
#include <math.h>

typedef __attribute__((ext_vector_type(2))) float v2f;
typedef __attribute__((ext_vector_type(8))) float v8f;

#define NN      50000
#define KNBR    32
#define LATD    10
#define NTILES  13      // ceil(200/16)
#define KCH     50      // split-K chunks
#define KSPAN   1000    // 50000 / 50
#define ZSTRIDE 208     // padded N for partial buffer

// ---------------------------------------------------------------------------
// Stage A: z_pre partials = x @ enc1_w.T  via V_WMMA_F32_16X16X4_F32 split-K.
// One wave per block. A = x tile (16x4, rows 4..15 zero), B = enc1_w.T tile
// (4x16). Deterministic: each (ntile,kchunk) block owns its partial slab.
// ---------------------------------------------------------------------------
__global__ __launch_bounds__(32)
void enc1_wmma_kernel(const float* __restrict__ x,
                      const float* __restrict__ w,        // enc1_w [200][50000]
                      float* __restrict__ part /* [KCH][4][ZSTRIDE] */) {
  const int lane  = threadIdx.x;
  const int hi    = (lane >= 16) ? 1 : 0;
  const int l16   = lane & 15;
  const int ntile = blockIdx.x;
  const int kc    = blockIdx.y;

  // B column (enc1 output neuron) handled by this lane; clamp + mask (no EXEC
  // divergence at the WMMA).
  const int   ncol   = ntile * 16 + l16;
  const int   ncol_c = (ncol < 200) ? ncol : 199;
  const float nmask  = (ncol < 200) ? 1.0f : 0.0f;
  // A row (batch); only 0..3 live.
  const int   mrow   = (l16 < 4) ? l16 : 0;
  const float amask  = (l16 < 4) ? 1.0f : 0.0f;

  const long kofs = (long)kc * KSPAN + (hi ? 2 : 0);
  const float* xr = x + (long)mrow   * NN + kofs;   // A: K=+0,+1 (lo) / +2,+3 (hi)
  const float* wr = w + (long)ncol_c * NN + kofs;   // B: same K split

  v8f c = {0.f, 0.f, 0.f, 0.f, 0.f, 0.f, 0.f, 0.f};
  #pragma unroll 4
  for (int it = 0; it < KSPAN / 4; ++it) {
    v2f a, b;
    a.x = xr[0] * amask;  a.y = xr[1] * amask;
    b.x = wr[0] * nmask;  b.y = wr[1] * nmask;
    xr += 4;  wr += 4;
    c = __builtin_amdgcn_wmma_f32_16x16x4_f32(false, a, false, b,
                                              (short)0, c, false, false);
  }

  // C/D layout: VGPR r holds (M=r, N=lane) for lanes 0-15. Batch rows 0..3
  // live in c[0..3] of the low half-wave.
  if (!hi && ncol < ZSTRIDE) {
    #pragma unroll
    for (int r = 0; r < 4; ++r)
      part[((long)kc * 4 + r) * ZSTRIDE + ncol] = c[r];
  }
}

// ---------------------------------------------------------------------------
// Stage B: deterministic split-K reduction + silu + the whole tiny MLP chain.
// Single 256-thread block, everything in LDS.
// ---------------------------------------------------------------------------
__global__ __launch_bounds__(256)
void mlp_kernel(const float* __restrict__ part,
                const float* __restrict__ e1b,
                const float* __restrict__ e2w, const float* __restrict__ e2b,
                const float* __restrict__ h1w, const float* __restrict__ h1b,
                const float* __restrict__ h2w, const float* __restrict__ h2b,
                const float* __restrict__ h3w, const float* __restrict__ h3b,
                float* __restrict__ e_out /* [4][10] */,
                float* __restrict__ h_out /* [4][256] */) {
  __shared__ float z[4][200];
  __shared__ float e[4][LATD];
  __shared__ float hA[4][64];
  __shared__ float hB[4][64];
  const int tid = threadIdx.x;

  // reduce partials over KCH chunks (fixed order -> deterministic) + silu
  for (int i = tid; i < 4 * 200; i += 256) {
    const int b = i / 200, n = i % 200;
    float acc = 0.f;
    for (int kc = 0; kc < KCH; ++kc)
      acc += part[((long)kc * 4 + b) * ZSTRIDE + n];
    acc += e1b[n];
    z[b][n] = acc / (1.0f + expf(-acc));            // silu
  }
  __syncthreads();

  if (tid < 4 * LATD) {                              // e = z @ enc2_w.T + b
    const int b = tid / LATD, l = tid % LATD;
    float acc = e2b[l];
    for (int n = 0; n < 200; ++n) acc += z[b][n] * e2w[l * 200 + n];
    e[b][l] = acc;
    e_out[b * LATD + l] = acc;
  }
  __syncthreads();

  if (tid < 4 * 64) {                                // h1 = silu(e @ h1_w.T + b)
    const int b = tid / 64, o = tid % 64;
    float acc = h1b[o];
    #pragma unroll
    for (int l = 0; l < LATD; ++l) acc += e[b][l] * h1w[o * LATD + l];
    hA[b][o] = acc / (1.0f + expf(-acc));
  }
  __syncthreads();

  if (tid < 4 * 64) {                                // h2 = silu(h1 @ h2_w.T + b)
    const int b = tid / 64, o = tid % 64;
    float acc = h2b[o];
    for (int i2 = 0; i2 < 64; ++i2) acc += hA[b][i2] * h2w[o * 64 + i2];
    hB[b][o] = acc / (1.0f + expf(-acc));
  }
  __syncthreads();

  for (int i = tid; i < 4 * 256; i += 256) {         // sigmoid(0.005 * h3)
    const int b = i / 256, o = i % 256;
    float acc = h3b[o];
    for (int i2 = 0; i2 < 64; ++i2) acc += hB[b][i2] * h3w[o * 64 + i2];
    acc *= 0.005f;
    h_out[b * 256 + o] = 1.0f / (1.0f + expf(-acc));
  }
}

// ---------------------------------------------------------------------------
// Stage C: RBF window decode. One thread per (batch, point).
// ---------------------------------------------------------------------------
__global__ __launch_bounds__(128)
void decode_kernel(const float* __restrict__ e    /* [4][10]  */,
                   const float* __restrict__ hsig /* [4][256] */,
                   const int*   __restrict__ labels,
                   const int*   __restrict__ nid,
                   const float* __restrict__ ndist,
                   const float* __restrict__ dec_w /* [50000][10] */,
                   const float* __restrict__ dec_b,
                   const float* __restrict__ Bp,
                   float* __restrict__ out /* [4][50000] */) {
  const int j = blockIdx.x * blockDim.x + threadIdx.x;
  const int b = blockIdx.y;
  if (j >= NN) return;

  const float B0v  = Bp[0];
  const float Hb   = hsig[b * 256 + labels[j]];
  const float base = 1.0f - 0.5f * Hb;

  float iw[LATD];                                   // 1 / (B0*(1-H/2)^(l+1))^2
  float pw = 1.0f;
  #pragma unroll
  for (int l = 0; l < LATD; ++l) {
    pw *= base;
    const float wl = pw * B0v;                      // MU == 1
    iw[l] = 1.0f / (wl * wl);
  }

  float s[LATD], t[LATD];
  #pragma unroll
  for (int l = 0; l < LATD; ++l) { s[l] = 0.f; t[l] = 0.f; }

  const int*   nrow = nid   + (long)j * KNBR;
  const float* drow = ndist + (long)j * KNBR;
  for (int k = 0; k < KNBR; ++k) {
    const int   nk = nrow[k];
    const float d  = drow[k];
    const float d2 = d * d;
    const float* bas = dec_w + (long)nk * LATD;     // contiguous 10-float gather
    #pragma unroll
    for (int l = 0; l < LATD; ++l) {
      float win = fmaf(-d2, iw[l], 1.0f);
      win = win > 0.f ? win : 0.f;
      s[l] += win;
      t[l] += win * bas[l];
    }
  }

  float acc = dec_b[j];
  #pragma unroll
  for (int l = 0; l < LATD; ++l)
    acc += e[b * LATD + l] * (t[l] / s[l]);
  out[(long)b * NN + j] = acc;
}

// ---------------------------------------------------------------------------
extern "C" void kernel_launch(void* const* d_in, const int* in_sizes, int n_in,
                              void* d_out, int out_size, void* d_ws, size_t ws_size,
                              hipStream_t stream) {
  const float* x      = (const float*)d_in[0];
  const int*   nid    = (const int*)  d_in[1];
  const float* ndist  = (const float*)d_in[2];
  const int*   labels = (const int*)  d_in[3];
  const float* e1w    = (const float*)d_in[4];
  const float* e1b    = (const float*)d_in[5];
  const float* e2w    = (const float*)d_in[6];
  const float* e2b    = (const float*)d_in[7];
  const float* dw     = (const float*)d_in[8];
  const float* db     = (const float*)d_in[9];
  const float* h1w    = (const float*)d_in[10];
  const float* h1b    = (const float*)d_in[11];
  const float* h2w    = (const float*)d_in[12];
  const float* h2b    = (const float*)d_in[13];
  const float* h3w    = (const float*)d_in[14];
  const float* h3b    = (const float*)d_in[15];
  const float* Bp     = (const float*)d_in[16];
  float* out = (float*)d_out;

  float* ws      = (float*)d_ws;
  float* ws_part = ws;                              // KCH*4*ZSTRIDE floats
  float* ws_e    = ws_part + (long)KCH * 4 * ZSTRIDE;
  float* ws_h    = ws_e + 64;                       // 4*256 floats

  enc1_wmma_kernel<<<dim3(NTILES, KCH), 32, 0, stream>>>(x, e1w, ws_part);
  mlp_kernel<<<1, 256, 0, stream>>>(ws_part, e1b, e2w, e2b,
                                    h1w, h1b, h2w, h2b, h3w, h3b,
                                    ws_e, ws_h);
  decode_kernel<<<dim3((NN + 127) / 128, 4), 128, 0, stream>>>(
      ws_e, ws_h, labels, nid, ndist, dw, db, Bp, out);
}